// AdaptiveContrastiveLoss_65781719105622
// MI455X (gfx1250) — compile-verified
//
#include <hip/hip_runtime.h>

// ---------------------------------------------------------------------------
// AdaptiveContrastiveLoss for MI455X (gfx1250, wave32, WMMA)
//   x: [4096, 512] fp32 -> scalar loss
// Pipeline:
//   1) normalize rows -> xn (ws)
//   2) fp32 WMMA GEMM sim = xn @ xn^T (ws), fused pass-0 radix histogram
//   3) 3 more 8-bit radix-select passes (both thresholds at once, L2-resident)
//   4) masked sums/counts with deterministic two-stage reduction
// ---------------------------------------------------------------------------

typedef float v2f __attribute__((ext_vector_type(2)));
typedef float v8f __attribute__((ext_vector_type(8)));

#define ACL_N     4096u
#define ACL_D     512u
#define ACL_NTOT  (ACL_N * ACL_N)          // 16777216
#define ACL_TPOS  13421773u                // ceil(0.8 * N*N)  (0-based sorted index)
#define ACL_TNEG  3355444u                 // ceil(0.2 * N*N)
#define ACL_NB_LOSS 2048

struct AclSelState {
  unsigned prefix_pos;
  unsigned prefix_neg;
  unsigned t_pos;   // remaining rank within current prefix bucket
  unsigned t_neg;
};

// float bits -> monotonically ordered unsigned key
__device__ __forceinline__ unsigned acl_key(unsigned u) {
  return u ^ ((unsigned)((int)u >> 31) | 0x80000000u);
}
__device__ __forceinline__ unsigned acl_fkey(float f) {
  return acl_key(__float_as_uint(f));
}

// Wave32 ballot-aggregated LDS histogram add: one atomic per distinct bin
// per wave instead of one per lane. Handles heavily skewed distributions.
__device__ __forceinline__ void acl_hist_add(unsigned* hist, unsigned bin, bool pred) {
  bool done = !pred;
  for (;;) {
    unsigned long long alive = __ballot(!done);
    if (alive == 0ull) break;
    int leader = __builtin_ctzll(alive);
    unsigned lbin = (unsigned)__shfl((int)bin, leader, 32);
    bool take = (!done) && (bin == lbin);
    unsigned long long grp = __ballot(take);
    if (take && (int)__lane_id() == leader)
      atomicAdd(hist + lbin, (unsigned)__popcll(grp));
    if (take) done = true;
  }
}

__device__ __forceinline__ float acl_wred_f(float v) {
  #pragma unroll
  for (int off = 16; off > 0; off >>= 1) v += __shfl_down(v, off, 32);
  return v;
}
__device__ __forceinline__ unsigned acl_wred_u(unsigned v) {
  #pragma unroll
  for (int off = 16; off > 0; off >>= 1) v += (unsigned)__shfl_down((int)v, off, 32);
  return v;
}

// ---------------------------------------------------------------------------
__global__ __launch_bounds__(256) void acl_init(unsigned* __restrict__ h0,
                                                unsigned* __restrict__ hp,
                                                unsigned* __restrict__ hn,
                                                AclSelState* __restrict__ st) {
  unsigned t = threadIdx.x;
  if (t < 256) { h0[t] = 0u; hp[t] = 0u; hn[t] = 0u; }
  if (t == 0) {
    st->prefix_pos = 0u; st->prefix_neg = 0u;
    st->t_pos = ACL_TPOS; st->t_neg = ACL_TNEG;
  }
}

// ---------------------------------------------------------------------------
__global__ __launch_bounds__(256) void acl_normalize(const float* __restrict__ x,
                                                     float* __restrict__ xn) {
  const unsigned row = blockIdx.x;
  const unsigned tid = threadIdx.x;
  v2f v = *(const v2f*)(x + row * ACL_D + tid * 2u);
  float s = v.x * v.x + v.y * v.y;
  s = acl_wred_f(s);
  __shared__ float wsum[8];
  __shared__ float sscale;
  if ((tid & 31u) == 0u) wsum[tid >> 5] = s;
  __syncthreads();
  if (tid == 0) {
    float t = 0.f;
    #pragma unroll
    for (int i = 0; i < 8; ++i) t += wsum[i];
    sscale = 1.0f / fmaxf(sqrtf(t), 1e-8f);
  }
  __syncthreads();
  float sc = sscale;
  v.x *= sc; v.y *= sc;
  *(v2f*)(xn + row * ACL_D + tid * 2u) = v;
}

// ---------------------------------------------------------------------------
// fp32 WMMA GEMM: each block computes a 128x128 tile of sim; 8 waves, each
// wave owns a 64x32 sub-tile = 4x2 accumulators of 16x16.
// Epilogue stores sim and builds the pass-0 (top byte) radix histogram.
__global__ __launch_bounds__(256) void acl_gemm_hist(const float* __restrict__ xn,
                                                     float* __restrict__ sim,
                                                     unsigned* __restrict__ gh0) {
  __shared__ unsigned lh[256];
  if (threadIdx.x < 256) lh[threadIdx.x] = 0u;
  __syncthreads();

  const unsigned tid  = threadIdx.x;
  const unsigned w    = tid >> 5;
  const unsigned lane = tid & 31u;
  const unsigned half = lane >> 4;      // K-group select for f32 WMMA frags
  const unsigned m    = lane & 15u;

  const unsigned by = blockIdx.x >> 5;  // 32x32 grid of 128x128 tiles
  const unsigned bx = blockIdx.x & 31u;
  const unsigned i0 = by * 128u + (w & 1u) * 64u;
  const unsigned j0 = bx * 128u + (w >> 1) * 32u;

  v8f acc[4][2];
  #pragma unroll
  for (int ri = 0; ri < 4; ++ri)
    #pragma unroll
    for (int ci = 0; ci < 2; ++ci) { v8f z = {}; acc[ri][ci] = z; }

  // A fragment lane layout (16x4 f32): a[v] = A[m][k + 2*half + v]
  // B fragment (4x16) mirrored: b[v] = B[k + 2*half + v][m] = xn[j0+..+m][k+..]
  const float* arow[4];
  const float* brow[2];
  #pragma unroll
  for (int ri = 0; ri < 4; ++ri)
    arow[ri] = xn + (i0 + (unsigned)ri * 16u + m) * ACL_D + 2u * half;
  #pragma unroll
  for (int ci = 0; ci < 2; ++ci)
    brow[ci] = xn + (j0 + (unsigned)ci * 16u + m) * ACL_D + 2u * half;

  for (unsigned k = 0; k < ACL_D; k += 4u) {
    v2f a[4], b[2];
    #pragma unroll
    for (int ri = 0; ri < 4; ++ri) a[ri] = *(const v2f*)(arow[ri] + k);
    #pragma unroll
    for (int ci = 0; ci < 2; ++ci) b[ci] = *(const v2f*)(brow[ci] + k);
    #pragma unroll
    for (int ri = 0; ri < 4; ++ri)
      #pragma unroll
      for (int ci = 0; ci < 2; ++ci)
        acc[ri][ci] = __builtin_amdgcn_wmma_f32_16x16x4_f32(
            false, a[ri], false, b[ci], (short)0, acc[ri][ci], false, false);
  }

  // C/D layout: element v of acc lives at row = base + v + 8*half, col = base + m
  #pragma unroll
  for (int ri = 0; ri < 4; ++ri) {
    #pragma unroll
    for (int ci = 0; ci < 2; ++ci) {
      #pragma unroll
      for (int v = 0; v < 8; ++v) {
        float s = acc[ri][ci][v];
        unsigned row = i0 + (unsigned)ri * 16u + (unsigned)v + 8u * half;
        unsigned col = j0 + (unsigned)ci * 16u + m;
        sim[row * ACL_N + col] = s;                  // RT store: keep in L2
        acl_hist_add(lh, acl_fkey(s) >> 24, true);   // fused radix pass 0
      }
    }
  }
  __syncthreads();
  if (tid < 256) {
    unsigned c = lh[tid];
    if (c) atomicAdd(gh0 + tid, c);
  }
}

// ---------------------------------------------------------------------------
// Radix-select histogram pass over sim for both thresholds (passes 1..3).
__global__ __launch_bounds__(256) void acl_histpass(const uint4* __restrict__ keys,
                                                    unsigned n4,
                                                    const AclSelState* __restrict__ st,
                                                    unsigned mask, unsigned shift,
                                                    unsigned* __restrict__ gp,
                                                    unsigned* __restrict__ gn) {
  __shared__ unsigned hp[256], hn[256];
  if (threadIdx.x < 256) { hp[threadIdx.x] = 0u; hn[threadIdx.x] = 0u; }
  __syncthreads();
  const unsigned ppos = st->prefix_pos;
  const unsigned pneg = st->prefix_neg;
  unsigned idx = blockIdx.x * blockDim.x + threadIdx.x;
  const unsigned stride = gridDim.x * blockDim.x;
  for (; idx < n4; idx += stride) {
    uint4 v = keys[idx];
    unsigned u[4] = {v.x, v.y, v.z, v.w};
    #pragma unroll
    for (int c = 0; c < 4; ++c) {
      unsigned key = acl_key(u[c]);
      unsigned bin = (key >> shift) & 255u;
      acl_hist_add(hp, bin, (key & mask) == ppos);
      acl_hist_add(hn, bin, (key & mask) == pneg);
    }
  }
  __syncthreads();
  if (threadIdx.x < 256) {
    unsigned c0 = hp[threadIdx.x]; if (c0) atomicAdd(gp + threadIdx.x, c0);
    unsigned c1 = hn[threadIdx.x]; if (c1) atomicAdd(gn + threadIdx.x, c1);
  }
}

// ---------------------------------------------------------------------------
// Consume a pass's histograms: pick digit containing the target rank for each
// threshold, update prefix/rank, then zero the histograms for reuse.
__global__ __launch_bounds__(256) void acl_scan(unsigned* __restrict__ hp,
                                                unsigned* __restrict__ hn,
                                                unsigned shift,
                                                AclSelState* __restrict__ st) {
  const unsigned tid = threadIdx.x;
  if (tid == 0) {
    unsigned t = st->t_pos, cum = 0u; int d = 0;
    for (; d < 256; ++d) { unsigned c = hp[d]; if (cum + c > t) break; cum += c; }
    if (d == 256) d = 255;
    st->prefix_pos |= ((unsigned)d) << shift;
    st->t_pos = t - cum;
  } else if (tid == 1) {
    unsigned t = st->t_neg, cum = 0u; int d = 0;
    for (; d < 256; ++d) { unsigned c = hn[d]; if (cum + c > t) break; cum += c; }
    if (d == 256) d = 255;
    st->prefix_neg |= ((unsigned)d) << shift;
    st->t_neg = t - cum;
  }
  __syncthreads();
  if (tid < 256) { hp[tid] = 0u; hn[tid] = 0u; }
}

// ---------------------------------------------------------------------------
// Masked sums + counts with per-block partials (deterministic reduction).
__global__ __launch_bounds__(256) void acl_loss(const float4* __restrict__ sim4,
                                                unsigned n4,
                                                const AclSelState* __restrict__ st,
                                                float* __restrict__ pps,
                                                float* __restrict__ pns,
                                                unsigned* __restrict__ ppc,
                                                unsigned* __restrict__ pnc) {
  const unsigned kp = st->prefix_pos;   // exact 32-bit key of pos threshold
  const unsigned kn = st->prefix_neg;
  float ps = 0.f, ns = 0.f;
  unsigned pc = 0u, nc = 0u;
  unsigned idx = blockIdx.x * blockDim.x + threadIdx.x;
  const unsigned stride = gridDim.x * blockDim.x;
  for (; idx < n4; idx += stride) {
    float4 v = sim4[idx];
    float s4[4] = {v.x, v.y, v.z, v.w};
    #pragma unroll
    for (int c = 0; c < 4; ++c) {
      float s = s4[c];
      unsigned key = acl_fkey(s);
      if (key > kp) { ps += fmaxf(1.0f - s, 0.0f); ++pc; }
      if (key < kn) { ns += fmaxf(s, 0.0f);        ++nc; }
    }
  }
  ps = acl_wred_f(ps); ns = acl_wred_f(ns);
  pc = acl_wred_u(pc); nc = acl_wred_u(nc);
  __shared__ float sps[8], sns[8];
  __shared__ unsigned spc[8], snc[8];
  const unsigned w = threadIdx.x >> 5;
  if ((threadIdx.x & 31u) == 0u) { sps[w] = ps; sns[w] = ns; spc[w] = pc; snc[w] = nc; }
  __syncthreads();
  if (threadIdx.x == 0) {
    float a = 0.f, b = 0.f; unsigned c = 0u, d = 0u;
    #pragma unroll
    for (int i = 0; i < 8; ++i) { a += sps[i]; b += sns[i]; c += spc[i]; d += snc[i]; }
    pps[blockIdx.x] = a; pns[blockIdx.x] = b; ppc[blockIdx.x] = c; pnc[blockIdx.x] = d;
  }
}

// ---------------------------------------------------------------------------
__global__ void acl_finalize(const float* __restrict__ pps,
                             const float* __restrict__ pns,
                             const unsigned* __restrict__ ppc,
                             const unsigned* __restrict__ pnc,
                             int nblocks, float* __restrict__ out) {
  if (blockIdx.x == 0 && threadIdx.x == 0) {
    float ps = 0.f, ns = 0.f; unsigned pc = 0u, nc = 0u;
    for (int i = 0; i < nblocks; ++i) {   // fixed order -> deterministic
      ps += pps[i]; ns += pns[i]; pc += ppc[i]; nc += pnc[i];
    }
    out[0] = ps / (float)pc + ns / (float)nc;   // LOSS_WEIGHT = 1.0
  }
}

// ---------------------------------------------------------------------------
extern "C" void kernel_launch(void* const* d_in, const int* in_sizes, int n_in,
                              void* d_out, int out_size, void* d_ws, size_t ws_size,
                              hipStream_t stream) {
  (void)in_sizes; (void)n_in; (void)out_size; (void)ws_size;
  const float* x = (const float*)d_in[0];
  float* out = (float*)d_out;

  char* ws = (char*)d_ws;
  const size_t XN_BYTES  = (size_t)ACL_N * ACL_D * 4u;   // 8 MiB
  const size_t SIM_BYTES = (size_t)ACL_N * ACL_N * 4u;   // 64 MiB
  float*    xn    = (float*)ws;
  float*    sim   = (float*)(ws + XN_BYTES);
  unsigned* h0    = (unsigned*)(ws + XN_BYTES + SIM_BYTES);
  unsigned* hp    = h0 + 256;
  unsigned* hn    = hp + 256;
  AclSelState* st = (AclSelState*)(hn + 256);
  float*    pps   = (float*)(st + 1);
  float*    pns   = pps + ACL_NB_LOSS;
  unsigned* ppc   = (unsigned*)(pns + ACL_NB_LOSS);
  unsigned* pnc   = ppc + ACL_NB_LOSS;

  const unsigned n4 = ACL_NTOT / 4u;

  acl_init<<<1, 256, 0, stream>>>(h0, hp, hn, st);
  acl_normalize<<<ACL_N, 256, 0, stream>>>(x, xn);
  acl_gemm_hist<<<(ACL_N / 128) * (ACL_N / 128), 256, 0, stream>>>(xn, sim, h0);
  acl_scan<<<1, 256, 0, stream>>>(h0, h0, 24u, st);                     // pass 0
  acl_histpass<<<2048, 256, 0, stream>>>((const uint4*)sim, n4, st,
                                         0xFF000000u, 16u, hp, hn);     // pass 1
  acl_scan<<<1, 256, 0, stream>>>(hp, hn, 16u, st);
  acl_histpass<<<2048, 256, 0, stream>>>((const uint4*)sim, n4, st,
                                         0xFFFF0000u, 8u, hp, hn);      // pass 2
  acl_scan<<<1, 256, 0, stream>>>(hp, hn, 8u, st);
  acl_histpass<<<2048, 256, 0, stream>>>((const uint4*)sim, n4, st,
                                         0xFFFFFF00u, 0u, hp, hn);      // pass 3
  acl_scan<<<1, 256, 0, stream>>>(hp, hn, 0u, st);
  acl_loss<<<ACL_NB_LOSS, 256, 0, stream>>>((const float4*)sim, n4, st,
                                            pps, pns, ppc, pnc);
  acl_finalize<<<1, 1, 0, stream>>>(pps, pns, ppc, pnc, ACL_NB_LOSS, out);
}